// ContrastiveNetWithGAT_22041772163476
// MI455X (gfx1250) — compile-verified
//
#include <hip/hip_runtime.h>
#include <math.h>

#define N_NODES 10000
#define E_EDGES 320000
#define IN_DIM 512
#define HID_DIM 128
#define OUT_DIM 128
#define HEADS 4
#define GC 64
#define FDIM (HEADS * GC)   // 256
#define NEG_SLOPE 0.2f

typedef float v2f __attribute__((ext_vector_type(2)));
typedef float v8f __attribute__((ext_vector_type(8)));
typedef unsigned int v4u __attribute__((ext_vector_type(4)));
typedef int v4i __attribute__((ext_vector_type(4)));
typedef int v8i __attribute__((ext_vector_type(8)));

#if __has_builtin(__builtin_amdgcn_tensor_load_to_lds) && \
    __has_builtin(__builtin_amdgcn_s_wait_tensorcnt)
#define USE_TDM 1
#else
#define USE_TDM 0
#endif

// TDM LDS padding: 4 DWORDs inserted after every 256 DWORDs streamed.
// Matching read-side index transform (pairs at even offsets never straddle
// a 256-DWORD chunk, so base+1 is always valid for the second element).
__device__ __forceinline__ int lds_pad_idx(int idx) {
  return idx + ((idx >> 8) << 2);
}
// floats of LDS needed for a 16 x K f32 tile incl. padding (+ slack)
__host__ __device__ constexpr int smem_floats(int K) {
  return 16 * K + ((16 * K) >> 8) * 4 + 8;
}

#if USE_TDM
// Issue one TDM descriptor: load rows x width f32 tile (contiguous rows of
// `width`, row stride `width`) from gsrc into LDS at lds_byte_addr, with
// 4-DWORD padding every 256 DWORDs. Wave-level op; waits TENSORcnt before returning.
__device__ __forceinline__ void tdm_load_tile_f32(unsigned lds_byte_addr,
                                                  const float* gsrc,
                                                  int width, int rows) {
  unsigned long long ga = (unsigned long long)(uintptr_t)gsrc;
  v4u g0;
  g0[0] = 1u;                                   // count=1, is_restore=0
  g0[1] = lds_byte_addr;                        // lds_addr
  g0[2] = (unsigned)(ga & 0xffffffffull);       // global_addr[31:0]
  g0[3] = (unsigned)((ga >> 32) & 0x01ffffffull) | (2u << 30);  // type=2
  v8i g1;
  g1[0] = (2 << 16)                             // data_size = 4 bytes
        | (1 << 20)                             // pad_enable
        | (7 << 22)                             // pad_interval = 256 DWORDs
        | (3 << 25);                            // pad_amount  = 4 DWORDs
  g1[1] = (width & 0xffff) << 16;               // tensor_dim0[15:0]
  g1[2] = ((width >> 16) & 0xffff)              // tensor_dim0[31:16]
        | ((rows & 0xffff) << 16);              // tensor_dim1[15:0]
  g1[3] = (width & 0xffff) << 16;               // tile_dim0 (dim1 hi16 = 0)
  g1[4] = (rows & 0xffff);                      // tile_dim1 (tile_dim2 = 0)
  g1[5] = width;                                // tensor_dim0_stride[31:0]
  g1[6] = 0;
  g1[7] = 0;
  v4i g2 = {0, 0, 0, 0};
  v4i g3 = {0, 0, 0, 0};
  v8i g4 = {0, 0, 0, 0, 0, 0, 0, 0};            // clang-23 extended group
  __builtin_amdgcn_tensor_load_to_lds(g0, g1, g2, g3, g4, 0);
  __builtin_amdgcn_s_wait_tensorcnt(0);
}
#endif

// ---------------------------------------------------------------------------
// WMMA f32 GEMM: C[M,Ncols] = op(A[M,K] @ B[K,Ncols] + bias)
// One wave per 16x16 output tile (V_WMMA_F32_16X16X4_F32); A strip staged in
// LDS by the Tensor Data Mover (fallback: cooperative loads). M % 16 == 0.
// ---------------------------------------------------------------------------
template <bool RELU, bool BIAS>
__global__ void wmma_gemm_kernel(const float* __restrict__ A,
                                 const float* __restrict__ B,
                                 const float* __restrict__ bias,
                                 float* __restrict__ C,
                                 int K, int Ncols) {
  extern __shared__ float smem[];
  const int mtile = blockIdx.x;
  const float* Ablk = A + (size_t)mtile * 16 * K;

#if USE_TDM
  if (threadIdx.x < 32) {  // wave 0 drives the DMA, then waits TENSORcnt
    tdm_load_tile_f32(__builtin_amdgcn_groupstaticsize(), Ablk, K, 16);
  }
#else
  for (int idx = threadIdx.x; idx < 16 * K; idx += blockDim.x) {
    smem[lds_pad_idx(idx)] = Ablk[idx];
  }
#endif
  __syncthreads();

  const int wave = threadIdx.x >> 5;       // N-tile index
  const int lane = threadIdx.x & 31;
  const int l15  = lane & 15;              // A row / B-C column within tile
  const int hi   = lane >> 4;              // K split (0: K+0/1, 1: K+2/3)
  const int col  = wave * 16 + l15;        // global output column

  v8f acc = {};
  for (int k = 0; k < K; k += 4) {
    const int ka = k + 2 * hi;
    const int ai = lds_pad_idx(l15 * K + ka);
    v2f a, b;
    // A fragment 16x4: lanes 0-15 hold K=k..k+1, lanes 16-31 hold K=k+2..k+3
    a.x = smem[ai];
    a.y = smem[ai + 1];
    // B fragment 4x16: mirrored K split, columns striped over lanes
    b.x = B[(size_t)ka * Ncols + col];
    b.y = B[(size_t)(ka + 1) * Ncols + col];
    acc = __builtin_amdgcn_wmma_f32_16x16x4_f32(
        /*neg_a=*/false, a, /*neg_b=*/false, b,
        /*c_mod=*/(short)0, acc, /*reuse_a=*/false, /*reuse_b=*/false);
  }

  const float bv = BIAS ? bias[col] : 0.0f;
#pragma unroll
  for (int v = 0; v < 8; ++v) {
    // C/D layout: lanes 0-15 -> M=v, lanes 16-31 -> M=v+8
    float val = acc[v] + bv;
    if (RELU) val = fmaxf(val, 0.0f);
    C[(size_t)(mtile * 16 + v + 8 * hi) * Ncols + col] = val;
  }
}

// ---------------------------------------------------------------------------
// Per-node attention scores: a_src[n,h] = dot(g[n,h,:], att_src[h,:]), same dst
// ---------------------------------------------------------------------------
__global__ void attn_score_kernel(const float* __restrict__ g,
                                  const float* __restrict__ att_src,
                                  const float* __restrict__ att_dst,
                                  float* __restrict__ a_src,
                                  float* __restrict__ a_dst) {
  int t = blockIdx.x * blockDim.x + threadIdx.x;   // t = n*HEADS + h
  if (t >= N_NODES * HEADS) return;
  int n = t >> 2;
  int h = t & 3;
  const float* gp = g + (size_t)n * FDIM + h * GC;
  const float* as = att_src + h * GC;
  const float* ad = att_dst + h * GC;
  float ss = 0.0f, sd = 0.0f;
#pragma unroll 8
  for (int c = 0; c < GC; ++c) {
    float gv = gp[c];
    ss += gv * as[c];
    sd += gv * ad[c];
  }
  a_src[t] = ss;
  a_dst[t] = sd;
}

// ---------------------------------------------------------------------------
// Init: m = -inf, s = 0, out = broadcast(bias_g)
// ---------------------------------------------------------------------------
__global__ void init_kernel(float* __restrict__ m, float* __restrict__ s,
                            float* __restrict__ out,
                            const float* __restrict__ bias_g) {
  int i = blockIdx.x * blockDim.x + threadIdx.x;
  if (i < N_NODES * HEADS) {
    m[i] = -INFINITY;
    s[i] = 0.0f;
  }
  if (i < N_NODES * FDIM) out[i] = bias_g[i & (FDIM - 1)];
}

// ---------------------------------------------------------------------------
// Edge helpers
// ---------------------------------------------------------------------------
__device__ __forceinline__ void edge_sd(const int* __restrict__ ei, int t,
                                        int& s, int& d) {
  if (t < E_EDGES) {
    s = ei[t];
    d = ei[E_EDGES + t];
  } else {               // appended self-loops
    s = d = t - E_EDGES;
  }
}

__device__ __forceinline__ float leaky(float e) {
  return e > 0.0f ? e : NEG_SLOPE * e;
}

// Mixed-sign-correct float atomic max via native integer atomics
__device__ __forceinline__ void atomic_max_float(float* addr, float value) {
  if (value >= 0.0f)
    atomicMax((int*)addr, __float_as_int(value));
  else
    atomicMin((unsigned int*)addr, __float_as_uint(value));
}

// ---------------------------------------------------------------------------
// Pass 1: segment max of leaky-relu'd scores into m[dst,h]
// ---------------------------------------------------------------------------
__global__ void edge_max_kernel(const int* __restrict__ ei,
                                const float4* __restrict__ a_src4,
                                const float4* __restrict__ a_dst4,
                                float* __restrict__ m) {
  int t = blockIdx.x * blockDim.x + threadIdx.x;
  if (t >= E_EDGES + N_NODES) return;
  int s, d;
  edge_sd(ei, t, s, d);
  float4 as = a_src4[s];          // global_load_b128
  float4 ad = a_dst4[d];
  atomic_max_float(&m[d * HEADS + 0], leaky(as.x + ad.x));
  atomic_max_float(&m[d * HEADS + 1], leaky(as.y + ad.y));
  atomic_max_float(&m[d * HEADS + 2], leaky(as.z + ad.z));
  atomic_max_float(&m[d * HEADS + 3], leaky(as.w + ad.w));
}

// ---------------------------------------------------------------------------
// Pass 2: segment sum of exp(e - m[dst]) into s[dst,h]
// ---------------------------------------------------------------------------
__global__ void edge_sum_kernel(const int* __restrict__ ei,
                                const float4* __restrict__ a_src4,
                                const float4* __restrict__ a_dst4,
                                const float4* __restrict__ m4,
                                float* __restrict__ ssum) {
  int t = blockIdx.x * blockDim.x + threadIdx.x;
  if (t >= E_EDGES + N_NODES) return;
  int s, d;
  edge_sd(ei, t, s, d);
  float4 as = a_src4[s];
  float4 ad = a_dst4[d];
  float4 mv = m4[d];
  unsafeAtomicAdd(&ssum[d * HEADS + 0], __expf(leaky(as.x + ad.x) - mv.x));
  unsafeAtomicAdd(&ssum[d * HEADS + 1], __expf(leaky(as.y + ad.y) - mv.y));
  unsafeAtomicAdd(&ssum[d * HEADS + 2], __expf(leaky(as.z + ad.z) - mv.z));
  unsafeAtomicAdd(&ssum[d * HEADS + 3], __expf(leaky(as.w + ad.w) - mv.w));
}

// ---------------------------------------------------------------------------
// Pass 3: out[dst, h*GC+c] += alpha * g[src, h*GC+c]
// One 256-thread block per edge; per-thread native f32 atomic add (L2-resident)
// ---------------------------------------------------------------------------
__global__ void edge_agg_kernel(const int* __restrict__ ei,
                                const float* __restrict__ a_src,
                                const float* __restrict__ a_dst,
                                const float* __restrict__ m,
                                const float* __restrict__ ssum,
                                const float* __restrict__ g,
                                float* __restrict__ out) {
  int t = blockIdx.x;           // edge id
  int c = threadIdx.x;          // 0..255 channel
  int h = c >> 6;               // head
  int s, d;
  edge_sd(ei, t, s, d);
  float e = leaky(a_src[s * HEADS + h] + a_dst[d * HEADS + h]);
  float alpha = __expf(e - m[d * HEADS + h]) / ssum[d * HEADS + h];
  unsafeAtomicAdd(&out[(size_t)d * FDIM + c], alpha * g[(size_t)s * FDIM + c]);
}

// ---------------------------------------------------------------------------
// Launch pipeline
// ---------------------------------------------------------------------------
extern "C" void kernel_launch(void* const* d_in, const int* in_sizes, int n_in,
                              void* d_out, int out_size, void* d_ws,
                              size_t ws_size, hipStream_t stream) {
  const float* x       = (const float*)d_in[0];
  const int*   ei      = (const int*)d_in[1];
  const float* W1      = (const float*)d_in[2];
  const float* b1      = (const float*)d_in[3];
  const float* W2      = (const float*)d_in[4];
  const float* b2      = (const float*)d_in[5];
  const float* Wg      = (const float*)d_in[6];
  const float* att_src = (const float*)d_in[7];
  const float* att_dst = (const float*)d_in[8];
  const float* bias_g  = (const float*)d_in[9];
  float* out = (float*)d_out;

  float* ws    = (float*)d_ws;
  float* h1    = ws;                               // N*128
  float* h     = h1 + (size_t)N_NODES * HID_DIM;   // N*128
  float* g     = h + (size_t)N_NODES * OUT_DIM;    // N*256
  float* a_src = g + (size_t)N_NODES * FDIM;       // N*4
  float* a_dst = a_src + (size_t)N_NODES * HEADS;  // N*4
  float* mbuf  = a_dst + (size_t)N_NODES * HEADS;  // N*4
  float* sbuf  = mbuf + (size_t)N_NODES * HEADS;   // N*4

  const int mtiles = N_NODES / 16;  // 625 exactly

  // h1 = relu(x @ W1 + b1)
  wmma_gemm_kernel<true, true>
      <<<mtiles, 32 * (HID_DIM / 16), smem_floats(IN_DIM) * sizeof(float),
         stream>>>(x, W1, b1, h1, IN_DIM, HID_DIM);
  // h = h1 @ W2 + b2
  wmma_gemm_kernel<false, true>
      <<<mtiles, 32 * (OUT_DIM / 16), smem_floats(HID_DIM) * sizeof(float),
         stream>>>(h1, W2, b2, h, HID_DIM, OUT_DIM);
  // g = h @ Wg
  wmma_gemm_kernel<false, false>
      <<<mtiles, 32 * (FDIM / 16), smem_floats(OUT_DIM) * sizeof(float),
         stream>>>(h, Wg, nullptr, g, OUT_DIM, FDIM);

  attn_score_kernel<<<(N_NODES * HEADS + 255) / 256, 256, 0, stream>>>(
      g, att_src, att_dst, a_src, a_dst);

  init_kernel<<<(N_NODES * FDIM + 255) / 256, 256, 0, stream>>>(mbuf, sbuf, out,
                                                                bias_g);

  const int T = E_EDGES + N_NODES;
  edge_max_kernel<<<(T + 255) / 256, 256, 0, stream>>>(
      ei, (const float4*)a_src, (const float4*)a_dst, mbuf);
  edge_sum_kernel<<<(T + 255) / 256, 256, 0, stream>>>(
      ei, (const float4*)a_src, (const float4*)a_dst, (const float4*)mbuf,
      sbuf);
  edge_agg_kernel<<<T, FDIM, 0, stream>>>(ei, a_src, a_dst, mbuf, sbuf, g, out);
}